// HeroGATConv_72739566125588
// MI455X (gfx1250) — compile-verified
//
#include <hip/hip_runtime.h>
#include <hip/hip_bf16.h>
#include <math.h>

// ---------------------------------------------------------------------------
// HeroGATConv on gfx1250 (MI455X).
// GEMM h = feat @ W_fc.T done with V_WMMA_F32_16X16X4_F32 (fp32 matrix pipe).
// Edge softmax via order-preserving uint atomicMax + f32 atomicAdd.
// ---------------------------------------------------------------------------

typedef __attribute__((ext_vector_type(2))) float v2f;
typedef __attribute__((ext_vector_type(8))) float v8f;

#define IN_F 128
#define HD   256   // H*D
#define H2C  4     // H/2
#define DD   32
#define NEG_SLOPE 0.2f

// order-preserving float -> uint key (monotone: a<b  <=>  key(a)<key(b))
__device__ __forceinline__ unsigned fkey(float f) {
    unsigned u = __float_as_uint(f);
    return (u & 0x80000000u) ? ~u : (u | 0x80000000u);
}
__device__ __forceinline__ float keyf(unsigned k) {
    return (k & 0x80000000u) ? __uint_as_float(k ^ 0x80000000u)
                             : __uint_as_float(~k);
}
// fkey(-INFINITY) == ~0xFF800000 == 0x007FFFFF
#define NEGINF_KEY 0x007FFFFFu

// ---------------------------------------------------------------------------
// Kernel 1: h = feat @ W_fc.T  (n x 128) @ (128 x 256) using f32 WMMA 16x16x4.
// One block = 16 rows of feat staged in LDS, 16 waves each owning a 16-col tile.
// ---------------------------------------------------------------------------
__global__ __launch_bounds__(512) void gemm_fc_wmma(
    const float* __restrict__ feat, const float* __restrict__ Wfc,
    float* __restrict__ h, int n)
{
    __shared__ float As[16 * IN_F];
    const int row0 = blockIdx.x * 16;

    for (int i = threadIdx.x; i < 16 * IN_F; i += 512) {
        int r = i >> 7, c = i & 127;
        int gr = row0 + r;
        As[i] = (gr < n) ? feat[(size_t)gr * IN_F + c] : 0.0f;
    }
    __syncthreads();

    const int wave  = threadIdx.x >> 5;       // 0..15 -> column tile
    const int lane  = threadIdx.x & 31;
    const int m     = lane & 15;              // A row / B col within tile
    const int khalf = (lane >> 4) * 2;        // K sub-offset: 0 or 2
    const int col0  = wave * 16;

    const float* __restrict__ wrow = Wfc + (size_t)(col0 + m) * IN_F;
    const float* __restrict__ arow = As + m * IN_F;

    v8f acc = {};
    #pragma unroll 4
    for (int k0 = 0; k0 < IN_F; k0 += 4) {
        v2f a, b;
        a.x = arow[k0 + khalf];
        a.y = arow[k0 + khalf + 1];
        b.x = wrow[k0 + khalf];
        b.y = wrow[k0 + khalf + 1];
        acc = __builtin_amdgcn_wmma_f32_16x16x4_f32(
            false, a, false, b, (short)0, acc, false, false);
    }

    // D layout: VGPR r -> tile row r + (lane>>4)*8, col = lane&15
    const int rowoff = (lane >> 4) * 8;
    const int col    = col0 + (lane & 15);
    float* __restrict__ out = h + (size_t)(row0 + rowoff) * HD + col;

    if (row0 + 16 <= n) {                     // uniform fast path: whole tile valid
        #pragma unroll
        for (int r = 0; r < 8; ++r) out[(size_t)r * HD] = acc[r];
    } else {                                  // ragged tail (n not multiple of 16)
        #pragma unroll
        for (int r = 0; r < 8; ++r)
            if (row0 + rowoff + r < n) out[(size_t)r * HD] = acc[r];
    }
}

// ---------------------------------------------------------------------------
// Kernel 2: per-node attention scores el/er (pos heads) and eln/ern (neg heads)
// one thread per (node, h2)
// ---------------------------------------------------------------------------
__global__ void attn_scores(
    const float* __restrict__ h,
    const float* __restrict__ attn_l,  const float* __restrict__ attn_r,
    const float* __restrict__ attn_ln, const float* __restrict__ attn_rn,
    float* __restrict__ el, float* __restrict__ er,
    float* __restrict__ eln, float* __restrict__ ern, int n)
{
    int idx = blockIdx.x * blockDim.x + threadIdx.x;
    if (idx >= n * H2C) return;
    int node = idx >> 2, h2 = idx & 3;
    const float* ft  = h + (size_t)node * HD + h2 * DD;
    const float* ftn = ft + H2C * DD;
    float sl = 0.f, sr = 0.f, sln = 0.f, srn = 0.f;
    #pragma unroll
    for (int d = 0; d < DD; ++d) {
        float v = ft[d], vn = ftn[d];
        sl  += v  * attn_l [h2 * DD + d];
        sr  += v  * attn_r [h2 * DD + d];
        sln += vn * attn_ln[h2 * DD + d];
        srn += vn * attn_rn[h2 * DD + d];
    }
    el[idx] = sl; er[idx] = sr; eln[idx] = sln; ern[idx] = srn;
}

// ---------------------------------------------------------------------------
// Kernel 3: ftp = sigmoid( elu(alpha_hidden @ W1.T + b1) @ W2.T + b2 )
// one wave32 per node; lane = hidden dim (D=32)
// ---------------------------------------------------------------------------
__global__ __launch_bounds__(256) void alpha_mlp(
    const float* __restrict__ ah, const float* __restrict__ W1,
    const float* __restrict__ b1, const float* __restrict__ W2,
    const float* __restrict__ b2, float* __restrict__ ftp, int n)
{
    int gid  = blockIdx.x * blockDim.x + threadIdx.x;
    int node = gid >> 5;
    int lane = gid & 31;
    if (node >= n) return;
    const float* x = ah + (size_t)node * IN_F;
    const float* w = W1 + (size_t)lane * IN_F;
    float s = b1[lane];
    #pragma unroll 4
    for (int k = 0; k < IN_F; ++k) s += x[k] * w[k];
    float hid = (s > 0.f) ? s : (expf(s) - 1.f);      // elu
    float v = hid * W2[lane];
    #pragma unroll
    for (int off = 16; off > 0; off >>= 1) v += __shfl_down(v, off, 32);
    if (lane == 0) ftp[node] = 1.f / (1.f + expf(-(v + b2[0])));
}

// ---------------------------------------------------------------------------
// Kernel 4: init — rst := bias broadcast, max keys := key(-inf), denoms := 0
// ---------------------------------------------------------------------------
__global__ void init_state(
    float* __restrict__ rst, const float* __restrict__ bias,
    unsigned* __restrict__ maxpos, unsigned* __restrict__ maxneg,
    float* __restrict__ denpos, float* __restrict__ denneg, int n)
{
    int i = blockIdx.x * blockDim.x + threadIdx.x;
    if (i < n * HD) rst[i] = bias[i & (HD - 1)];
    if (i < n * H2C) {
        maxpos[i] = NEGINF_KEY; maxneg[i] = NEGINF_KEY;
        denpos[i] = 0.f;        denneg[i] = 0.f;
    }
}

// ---------------------------------------------------------------------------
// Kernel 5: edge pass 1 — lrelu scores, alpha gate, segment-max via atomicMax
// scores for the negative branch stored already negated (softmax(-en)).
// ---------------------------------------------------------------------------
__global__ void edge_pass1(
    const int* __restrict__ src, const int* __restrict__ dst,
    const float* __restrict__ el,  const float* __restrict__ er,
    const float* __restrict__ eln, const float* __restrict__ ern,
    const float* __restrict__ ftp,
    float* __restrict__ epos, float* __restrict__ eneg,
    unsigned* __restrict__ maxpos, unsigned* __restrict__ maxneg,
    float* __restrict__ alpha_out, int eN)
{
    int e = blockIdx.x * blockDim.x + threadIdx.x;
    if (e >= eN) return;
    int s = src[e], d = dst[e];
    #pragma unroll
    for (int hh = 0; hh < H2C; ++hh) {
        float x = el[s * H2C + hh] + er[d * H2C + hh];
        x = (x >= 0.f) ? x : NEG_SLOPE * x;
        epos[(size_t)e * H2C + hh] = x;
        atomicMax(&maxpos[d * H2C + hh], fkey(x));

        float y = eln[s * H2C + hh] + ern[d * H2C + hh];
        y = (y >= 0.f) ? y : NEG_SLOPE * y;
        y = -y;                                   // softmax over -en
        eneg[(size_t)e * H2C + hh] = y;
        atomicMax(&maxneg[d * H2C + hh], fkey(y));
    }
    alpha_out[e] = 1.f / (1.f + expf(-(ftp[s] + ftp[d])));
}

// ---------------------------------------------------------------------------
// Kernel 6: convert max keys back to floats, -inf -> 0 (empty segments)
// ---------------------------------------------------------------------------
__global__ void finalize_max(unsigned* __restrict__ maxpos,
                             unsigned* __restrict__ maxneg, int n4)
{
    int i = blockIdx.x * blockDim.x + threadIdx.x;
    if (i >= n4) return;
    float mp = keyf(maxpos[i]); if (!isfinite(mp)) mp = 0.f;
    float mn = keyf(maxneg[i]); if (!isfinite(mn)) mn = 0.f;
    ((float*)maxpos)[i] = mp;
    ((float*)maxneg)[i] = mn;
}

// ---------------------------------------------------------------------------
// Kernel 7: edge pass 2 — ex = exp(score - m[dst]); denom += ex (atomicAdd)
// overwrites score buffers with ex in place.
// ---------------------------------------------------------------------------
__global__ void edge_pass2(
    const int* __restrict__ dst,
    float* __restrict__ epos, float* __restrict__ eneg,
    const float* __restrict__ mpos, const float* __restrict__ mneg,
    float* __restrict__ denpos, float* __restrict__ denneg, int eN)
{
    int e = blockIdx.x * blockDim.x + threadIdx.x;
    if (e >= eN) return;
    int d = dst[e];
    #pragma unroll
    for (int hh = 0; hh < H2C; ++hh) {
        float exp_p = expf(epos[(size_t)e * H2C + hh] - mpos[d * H2C + hh]);
        epos[(size_t)e * H2C + hh] = exp_p;
        atomicAdd(&denpos[d * H2C + hh], exp_p);
        float exp_n = expf(eneg[(size_t)e * H2C + hh] - mneg[d * H2C + hh]);
        eneg[(size_t)e * H2C + hh] = exp_n;
        atomicAdd(&denneg[d * H2C + hh], exp_n);
    }
}

// ---------------------------------------------------------------------------
// Kernel 8: edge pass 3 — scatter-add ft[src]*a into rst. One wave per edge;
// lane = feature dim. 256 f32 atomicAdds per edge.
// ---------------------------------------------------------------------------
__global__ __launch_bounds__(256) void edge_pass3(
    const int* __restrict__ src, const int* __restrict__ dst,
    const float* __restrict__ h,
    const float* __restrict__ epos, const float* __restrict__ eneg,
    const float* __restrict__ denpos, const float* __restrict__ denneg,
    const float* __restrict__ alpha_e, float* __restrict__ rst, int eN)
{
    int gid  = blockIdx.x * blockDim.x + threadIdx.x;
    int e    = gid >> 5;
    int lane = gid & 31;
    if (e >= eN) return;
    int s = src[e], d = dst[e];
    float al = alpha_e[e];
    const float* hs  = h   + (size_t)s * HD;
    float*       out = rst + (size_t)d * HD;
    #pragma unroll
    for (int hh = 0; hh < H2C; ++hh) {
        float a = epos[(size_t)e * H2C + hh] / denpos[d * H2C + hh] * al;
        atomicAdd(&out[hh * DD + lane], hs[hh * DD + lane] * a);
        float an = eneg[(size_t)e * H2C + hh] / denneg[d * H2C + hh] * (1.f - al);
        atomicAdd(&out[H2C * DD + hh * DD + lane],
                  hs[H2C * DD + hh * DD + lane] * an);
    }
}

// ---------------------------------------------------------------------------
extern "C" void kernel_launch(void* const* d_in, const int* in_sizes, int n_in,
                              void* d_out, int out_size, void* d_ws, size_t ws_size,
                              hipStream_t stream) {
    const int*   src     = (const int*)  d_in[0];
    const int*   dst     = (const int*)  d_in[1];
    const float* feat    = (const float*)d_in[2];
    const float* ah      = (const float*)d_in[3];
    const float* Wfc     = (const float*)d_in[4];
    const float* attn_l  = (const float*)d_in[5];
    const float* attn_r  = (const float*)d_in[6];
    const float* attn_ln = (const float*)d_in[7];
    const float* attn_rn = (const float*)d_in[8];
    const float* bias    = (const float*)d_in[9];
    const float* W1      = (const float*)d_in[10];
    const float* b1      = (const float*)d_in[11];
    const float* W2      = (const float*)d_in[12];
    const float* b2      = (const float*)d_in[13];

    const int n  = in_sizes[2] / IN_F;   // 50000
    const int eN = in_sizes[0];          // 800000

    float* rst     = (float*)d_out;                    // n*256
    float* alpha_o = (float*)d_out + (size_t)n * HD;   // eN

    // workspace carve-up (floats)
    float* ws = (float*)d_ws;
    size_t off = 0;
    float*    hbuf   = ws + off; off += (size_t)n * HD;
    float*    el     = ws + off; off += (size_t)n * H2C;
    float*    er     = ws + off; off += (size_t)n * H2C;
    float*    eln    = ws + off; off += (size_t)n * H2C;
    float*    ern    = ws + off; off += (size_t)n * H2C;
    float*    ftp    = ws + off; off += (size_t)n;
    unsigned* maxpos = (unsigned*)(ws + off); off += (size_t)n * H2C;
    unsigned* maxneg = (unsigned*)(ws + off); off += (size_t)n * H2C;
    float*    denpos = ws + off; off += (size_t)n * H2C;
    float*    denneg = ws + off; off += (size_t)n * H2C;
    float*    epos   = ws + off; off += (size_t)eN * H2C;
    float*    eneg   = ws + off; off += (size_t)eN * H2C;

    const int nt = (n + 15) / 16;

    gemm_fc_wmma<<<nt, 512, 0, stream>>>(feat, Wfc, hbuf, n);

    attn_scores<<<(n * H2C + 255) / 256, 256, 0, stream>>>(
        hbuf, attn_l, attn_r, attn_ln, attn_rn, el, er, eln, ern, n);

    alpha_mlp<<<(n * 32 + 255) / 256, 256, 0, stream>>>(
        ah, W1, b1, W2, b2, ftp, n);

    init_state<<<((size_t)n * HD + 255) / 256, 256, 0, stream>>>(
        rst, bias, maxpos, maxneg, denpos, denneg, n);

    edge_pass1<<<(eN + 255) / 256, 256, 0, stream>>>(
        src, dst, el, er, eln, ern, ftp, epos, eneg, maxpos, maxneg, alpha_o, eN);

    finalize_max<<<(n * H2C + 255) / 256, 256, 0, stream>>>(maxpos, maxneg, n * H2C);

    edge_pass2<<<(eN + 255) / 256, 256, 0, stream>>>(
        dst, epos, eneg, (float*)maxpos, (float*)maxneg, denpos, denneg, eN);

    edge_pass3<<<((size_t)eN * 32 + 255) / 256, 256, 0, stream>>>(
        src, dst, hbuf, epos, eneg, denpos, denneg, alpha_o, rst, eN);
}